// ConsistencyLoss_22299470201498
// MI455X (gfx1250) — compile-verified
//
#include <hip/hip_runtime.h>
#include <hip/hip_bf16.h>
#include <math.h>

// KL(p || q) * T^2, batch-mean, C=5, T=3. Memory-bound streaming reduction.

typedef __attribute__((ext_vector_type(2))) float v2f;
typedef __attribute__((ext_vector_type(8))) float v8f;

#define NC   5
#define TINV 0.3333333333333333f   // 1/T, T = 3.0

// ---- 32-lane wave sum via V_WMMA_F32_16X16X4_F32 ------------------------
// A (16x4 f32, 2 VGPRs): lanes 0-15 put acc in A[m][0], lanes 16-31 in
// A[m][2]; second VGPR = 0. B = all ones (layout-independent). C = 0.
// D[m][*] = acc[m] + acc[m+16]. D VGPR v: lanes 0-15 = rowsum_v,
// lanes 16-31 = rowsum_{v+8}. Sum the 8 D VGPRs lane-wise, then fold the
// two half-waves with one shfl_xor(16). All lanes end with the full sum.
__device__ __forceinline__ float wave_sum_wmma(float acc) {
  v2f a; a[0] = acc;  a[1] = 0.0f;
  v2f b; b[0] = 1.0f; b[1] = 1.0f;
  v8f c = {};
  v8f d = __builtin_amdgcn_wmma_f32_16x16x4_f32(
      /*neg_a=*/false, a, /*neg_b=*/false, b,
      /*c_mod=*/(short)0, c, /*reuse_a=*/false, /*reuse_b=*/false);
  float s = ((d[0] + d[1]) + (d[2] + d[3])) + ((d[4] + d[5]) + (d[6] + d[7]));
  s += __shfl_xor(s, 16, 32);
  return s;
}

// ---- per-sample KL term --------------------------------------------------
__device__ __forceinline__ float kl_one(float s, const float* __restrict__ l) {
  // soft targets (vectorized form of the reference loop)
  float x      = fminf(fmaxf(s * 5.0f, 0.0f), 4.0f);
  int   idx    = (int)x;                         // floor (x >= 0)
  float center = ((float)idx + 0.5f) * 0.2f;
  float dist   = fabsf(s - center) * 5.0f;
  bool  lo     = (idx > 0) && (s < center);
  bool  hi     = (idx < NC - 1) && (s > center);
  bool  has    = lo || hi;
  float mainv  = has ? 1.0f - dist : 1.0f;
  int   nbi    = idx + (hi ? 1 : 0) - (lo ? 1 : 0);
  float nbv    = has ? dist : 0.0f;

  // p = softmax(soft/T); log p = t - logZ
  float t[NC], e[NC];
  float Z = 0.0f;
#pragma unroll
  for (int c = 0; c < NC; ++c) {
    float soft = (c == idx ? mainv : 0.0f) + (c == nbi ? nbv : 0.0f);
    t[c] = soft * TINV;
    e[c] = __expf(t[c]);
    Z += e[c];
  }
  float logZ = __logf(Z);
  float rZ   = 1.0f / Z;

  // log q = log_softmax(logits/T) with max-subtraction
  float u[NC];
  float mu = -3.4e38f;
#pragma unroll
  for (int c = 0; c < NC; ++c) { u[c] = l[c] * TINV; mu = fmaxf(mu, u[c]); }
  float ze = 0.0f;
#pragma unroll
  for (int c = 0; c < NC; ++c) ze += __expf(u[c] - mu);
  float logze = __logf(ze);

  float kl = 0.0f;
#pragma unroll
  for (int c = 0; c < NC; ++c) {
    float p    = e[c] * rZ;
    float logp = t[c] - logZ;
    float logq = u[c] - mu - logze;
    kl += p * (logp - logq);
  }
  return kl;
}

// ---- pass 1: grid-stride streaming, per-block partial sums ---------------
__global__ __launch_bounds__(256)
void kl_partial_kernel(const float* __restrict__ qs, const float* __restrict__ lg,
                       float* __restrict__ partial, int n) {
  const long long tid    = (long long)blockIdx.x * blockDim.x + threadIdx.x;
  const long long stride = (long long)gridDim.x * blockDim.x;
  const long long n4     = (long long)(n >> 2);

  const float4* qs4 = (const float4*)qs;
  const float4* lg4 = (const float4*)lg;   // 4 samples = 5 aligned float4 (80B)

  float acc = 0.0f;
  for (long long g = tid; g < n4; g += stride) {
    float4 sc = qs4[g];
    float4 v0 = lg4[5 * g + 0];
    float4 v1 = lg4[5 * g + 1];
    float4 v2 = lg4[5 * g + 2];
    float4 v3 = lg4[5 * g + 3];
    float4 v4 = lg4[5 * g + 4];
    float r0[NC] = {v0.x, v0.y, v0.z, v0.w, v1.x};
    float r1[NC] = {v1.y, v1.z, v1.w, v2.x, v2.y};
    float r2[NC] = {v2.z, v2.w, v3.x, v3.y, v3.z};
    float r3[NC] = {v3.w, v4.x, v4.y, v4.z, v4.w};
    acc += kl_one(sc.x, r0);
    acc += kl_one(sc.y, r1);
    acc += kl_one(sc.z, r2);
    acc += kl_one(sc.w, r3);
  }
  // tail (n % 4), one sample per low thread
  long long base = n4 << 2;
  if (tid < (long long)n - base) {
    long long i = base + tid;
    float r[NC];
#pragma unroll
    for (int c = 0; c < NC; ++c) r[c] = lg[i * NC + c];
    acc += kl_one(qs[i], r);
  }

  // uniform control flow from here on: EXEC all-ones for WMMA
  float wsum = wave_sum_wmma(acc);

  __shared__ float smem[8];
  int lane = threadIdx.x & 31;
  int wid  = threadIdx.x >> 5;
  if (lane == 0) smem[wid] = wsum;
  __syncthreads();
  if (threadIdx.x == 0) {
    float b = 0.0f;
#pragma unroll
    for (int w = 0; w < 8; ++w) b += smem[w];
    partial[blockIdx.x] = b;
  }
}

// ---- pass 2: single block folds partials, writes the scalar --------------
__global__ __launch_bounds__(256)
void kl_final_kernel(const float* __restrict__ partial, int nparts,
                     float* __restrict__ out, float scale) {
  float acc = 0.0f;
  for (int i = threadIdx.x; i < nparts; i += 256) acc += partial[i];

  float wsum = wave_sum_wmma(acc);

  __shared__ float smem[8];
  int lane = threadIdx.x & 31;
  int wid  = threadIdx.x >> 5;
  if (lane == 0) smem[wid] = wsum;
  __syncthreads();
  if (threadIdx.x == 0) {
    float b = 0.0f;
#pragma unroll
    for (int w = 0; w < 8; ++w) b += smem[w];
    out[0] = b * scale;   // overwrite: deterministic across graph replays
  }
}

extern "C" void kernel_launch(void* const* d_in, const int* in_sizes, int n_in,
                              void* d_out, int out_size, void* d_ws, size_t ws_size,
                              hipStream_t stream) {
  const float* qs = (const float*)d_in[0];   // quality_score [B]
  const float* lg = (const float*)d_in[1];   // class_logits  [B,5]
  int n = in_sizes[0];

  float* out     = (float*)d_out;
  float* partial = (float*)d_ws;

  int nblocks = 2048;                        // 512K threads; 2 float4-groups each
  if ((size_t)nblocks * sizeof(float) > ws_size) {
    nblocks = (int)(ws_size / sizeof(float));
    if (nblocks < 1) nblocks = 1;
  }

  kl_partial_kernel<<<nblocks, 256, 0, stream>>>(qs, lg, partial, n);
  kl_final_kernel<<<1, 256, 0, stream>>>(partial, nblocks, out,
                                         9.0f / (float)n);  // T^2 / B
}